// BaseCouplingLayer_2499670966641
// MI455X (gfx1250) — compile-verified
//
#include <hip/hip_runtime.h>

// CDNA5 / gfx1250: wave32, WMMA. Exact-f32 matrix op: V_WMMA_F32_16X16X4_F32.
typedef __attribute__((ext_vector_type(2))) float v2f;
typedef __attribute__((ext_vector_type(8))) float v8f;

#define BDIM 262144
#define DDIM 256        // x row length
#define MDIM 128        // D/2
#define HDIM 256        // hidden
// Row-pair interleaved weight layouts: Wp[p][2*col + r] = W[2p+r][col], r in {0,1}.
// Pair-row strides chosen so stride % 64 == 32: lane half 0 b64 loads cover banks
// {c..c+31}, lane half 1 (+1 pair row) covers {c+32..c+63} -> conflict-free.
#define W1P_STRIDE 544  // 2*256 + 32  (byte stride 2176: 16B aligned)
#define W2P_STRIDE 288  // 2*128 + 32  (byte stride 1152: 16B aligned)
#define HS_STRIDE  20   // 16 + pad: even (b64-aligned), banks of lane halves disjoint

#define LDS_W1P   (64  * W1P_STRIDE)   // 34816 floats
#define LDS_W2P   (128 * W2P_STRIDE)   // 36864 floats
#define LDS_HS    (8 * 2 * 16 * HS_STRIDE) // 5120 floats (double-buffered per wave)
#define LDS_FLOATS (LDS_W1P + LDS_W2P + LDS_HS + HDIM + MDIM) // 77184
#define LDS_BYTES  (LDS_FLOATS * 4)    // 308736 <= 327680 (320KB WGP LDS)

__global__ __launch_bounds__(256)
void coupling_mlp_kernel(const float* __restrict__ x,
                         const float* __restrict__ W1,
                         const float* __restrict__ b1,
                         const float* __restrict__ W2,
                         const float* __restrict__ b2,
                         float* __restrict__ out)
{
    extern __shared__ float lds[];
    float* W1p = lds;                       // 64 pair-rows x 544
    float* W2p = W1p + LDS_W1P;             // 128 pair-rows x 288
    float* HSa = W2p + LDS_W2P;             // 8 waves x 2 bufs x (16 x 20)
    float* b1s = HSa + LDS_HS;              // 256
    float* b2s = b1s + HDIM;                // 128

    const int tid = threadIdx.x;

    // ---- Cooperative weight preload into pair-interleaved LDS layout ----
    // W1: 64 pair-rows x 128 col-pairs = 8192 float4 stores / 256 threads
#pragma unroll 4
    for (int i = 0; i < 32; ++i) {
        const int e = tid + i * 256;
        const int p = e >> 7, c = (e & 127) * 2;       // pair-row, even col
        const float2 g0 = *(const float2*)(W1 + (2 * p)     * HDIM + c);
        const float2 g1 = *(const float2*)(W1 + (2 * p + 1) * HDIM + c);
        float4 v; v.x = g0.x; v.y = g1.x; v.z = g0.y; v.w = g1.y;
        *(float4*)(W1p + p * W1P_STRIDE + 2 * c) = v;
    }
    // W2: 128 pair-rows x 64 col-pairs = 8192 float4 stores
#pragma unroll 4
    for (int i = 0; i < 32; ++i) {
        const int e = tid + i * 256;
        const int q = e >> 6, c = (e & 63) * 2;
        const float2 g0 = *(const float2*)(W2 + (2 * q)     * MDIM + c);
        const float2 g1 = *(const float2*)(W2 + (2 * q + 1) * MDIM + c);
        float4 v; v.x = g0.x; v.y = g1.x; v.z = g0.y; v.w = g1.y;
        *(float4*)(W2p + q * W2P_STRIDE + 2 * c) = v;
    }
    b1s[tid] = b1[tid];
    if (tid < MDIM) b2s[tid] = b2[tid];
    __syncthreads();

    const int lane = tid & 31;
    const int wave = tid >> 5;
    const int half = lane >> 4;     // 16-lane half
    const int n    = lane & 15;     // N index (B/C/D) == M index (A)
    float* HS0 = HSa + wave * (2 * 16 * HS_STRIDE);
    float* HS1 = HS0 + 16 * HS_STRIDE;

    const long r0 = ((long)blockIdx.x * 8 + wave) * 16;   // 16-row tile per wave

    // ---- GEMM1 A-fragments hoisted: one pass over this lane's x row ----
    // A 16x4 f32 layout: v0 = A[m][4k+2h], v1 = A[m][4k+2h+1]; first[:,c]=x[:,2c]
    const float* xrow = x + (size_t)(r0 + n) * DDIM;
    v2f ax[32];
#pragma unroll
    for (int k1 = 0; k1 < 32; ++k1) {
        const float4 t = *(const float4*)(xrow + 8 * k1 + 4 * half);
        v2f a; a.x = t.x; a.y = t.z;
        ax[k1] = a;
    }

    v8f acc[8];
#pragma unroll
    for (int t2 = 0; t2 < 8; ++t2) acc[t2] = {};

    // GEMM1: hidden tile t1 = relu(first @ W1[:,16t1:+16] + b1); stage into HSb.
    auto gemm1 = [&](int t1, float* HSb) {
        // pair-row p = 2*k1 + half ; col = 16*t1 + n
        const float* bptr = W1p + half * W1P_STRIDE + (16 * t1 + n) * 2;
        v2f bb[4];                                    // depth-4 load pipeline
#pragma unroll
        for (int j = 0; j < 4; ++j)
            bb[j] = *(const v2f*)(bptr + j * (2 * W1P_STRIDE));
        v8f h = {};
#pragma unroll
        for (int k1 = 0; k1 < 32; ++k1) {
            h = __builtin_amdgcn_wmma_f32_16x16x4_f32(false, ax[k1], false, bb[k1 & 3],
                                                      (short)0, h, false, false);
            if (k1 < 28)
                bb[k1 & 3] = *(const v2f*)(bptr + (k1 + 4) * (2 * W1P_STRIDE));
        }
        const float bias1 = b1s[16 * t1 + n];
#pragma unroll
        for (int j = 0; j < 8; ++j) {
            const float t = h[j] + bias1;
            h[j] = t > 0.f ? t : 0.f;
        }
        // D layout: VGPR j holds row (j + 8*half), col n
#pragma unroll
        for (int j = 0; j < 8; ++j)
            HSb[(j + 8 * half) * HS_STRIDE + n] = h[j];
    };

    // GEMM2 partial: acc += hidden_tile(t1) @ W2[16t1:+16, :]
    auto gemm2 = [&](int t1, const float* HSb) {
        const float* aptr = HSb + n * HS_STRIDE + 2 * half;      // A row m = n
        const float* bptr = W2p + (8 * t1 + half) * W2P_STRIDE + 2 * n;
        v2f a2 = *(const v2f*)(aptr);                            // k2 = 0
        v2f bbuf[8];
#pragma unroll
        for (int t2 = 0; t2 < 8; ++t2)
            bbuf[t2] = *(const v2f*)(bptr + 32 * t2);
#pragma unroll
        for (int k2 = 0; k2 < 4; ++k2) {
            v2f a2n = a2;
            if (k2 < 3) a2n = *(const v2f*)(aptr + 4 * (k2 + 1));
#pragma unroll
            for (int t2 = 0; t2 < 8; ++t2) {
                v2f bn = bbuf[t2];
                if (k2 < 3)
                    bn = *(const v2f*)(bptr + (k2 + 1) * (2 * W2P_STRIDE) + 32 * t2);
                acc[t2] = __builtin_amdgcn_wmma_f32_16x16x4_f32(false, a2, false, bbuf[t2],
                                                                (short)0, acc[t2], false, false);
                bbuf[t2] = bn;
            }
            a2 = a2n;
        }
    };

    // ---- Skewed pipeline: GEMM2(t1-1) overlaps GEMM1(t1); HS double-buffered ----
    float* bufA = HS0;   // holds tile t1-1
    float* bufB = HS1;   // being filled with tile t1
    gemm1(0, bufA);
#pragma unroll 1
    for (int t1 = 1; t1 < 16; ++t1) {
        gemm1(t1, bufB);
        gemm2(t1 - 1, bufA);
        float* tmp = bufA; bufA = bufB; bufB = tmp;
    }
    gemm2(15, bufA);

    // ---- Epilogue: out[:,2c]=first, out[:,2c+1]=second + m + b2 (8B stores) ----
#pragma unroll
    for (int t2 = 0; t2 < 8; ++t2) {
        const float bias2 = b2s[16 * t2 + n];
#pragma unroll
        for (int j = 0; j < 8; ++j) {
            const int  m   = j + 8 * half;                 // D-layout row
            const long idx = (r0 + m) * DDIM + 2 * (16 * t2 + n);
            const float2 xv = *(const float2*)(x + idx);   // {first, second}
            float2 o;
            o.x = xv.x;
            o.y = xv.y + acc[t2][j] + bias2;
            *(float2*)(out + idx) = o;
        }
    }
}

extern "C" void kernel_launch(void* const* d_in, const int* in_sizes, int n_in,
                              void* d_out, int out_size, void* d_ws, size_t ws_size,
                              hipStream_t stream) {
    const float* x  = (const float*)d_in[0];
    const float* W1 = (const float*)d_in[1];
    const float* b1 = (const float*)d_in[2];
    const float* W2 = (const float*)d_in[3];
    const float* b2 = (const float*)d_in[4];
    float* out = (float*)d_out;

    (void)hipFuncSetAttribute((const void*)coupling_mlp_kernel,
                              hipFuncAttributeMaxDynamicSharedMemorySize,
                              (int)LDS_BYTES);

    coupling_mlp_kernel<<<dim3(BDIM / (16 * 8)), dim3(256), LDS_BYTES, stream>>>(
        x, W1, b1, W2, b2, out);
}